// SpGraphAttentionLayer_1546188227120
// MI455X (gfx1250) — compile-verified
//
#include <hip/hip_runtime.h>
#include <hip/hip_bf16.h>
#include <math.h>

typedef float v2f __attribute__((ext_vector_type(2)));
typedef float v8f __attribute__((ext_vector_type(8)));

#define FDIM 128
#define MAXNORM 0.996f          // (1 - 4e-3)/sqrt(c), c=1
#define MIN_NORM 1e-15f
#define CLIP1 (1.0f - 1e-7f)

// ---------------------------------------------------------------- zero scratch
__global__ void kZero(float* __restrict__ p, size_t n) {
  size_t i = (size_t)blockIdx.x * blockDim.x + threadIdx.x;
  size_t stride = (size_t)gridDim.x * blockDim.x;
  for (; i < n; i += stride) p[i] = 0.0f;
}

// ---------------------------------------------------------------- WMMA GEMM
// mx = W @ x^T  (128 x N), stored NODE-major:  hT[n*128 + f] = sum_k W[f,k]*x[n,k]
// Also accumulates rowsumsq[f] = sum_n mx[f,n]^2  (for per-row scalar chain).
// Per block: 8 waves, 128 nodes; W staged in LDS (64KB).
__global__ __launch_bounds__(256) void kGemm(const float* __restrict__ x,
                                             const float* __restrict__ W,
                                             float* __restrict__ hT,
                                             float* __restrict__ rowsumsq,
                                             int N) {
  __shared__ float Wlds[FDIM * FDIM];  // 64 KB
  int tid = threadIdx.x;
  for (int i = tid * 4; i < FDIM * FDIM; i += 256 * 4)
    *(float4*)(Wlds + i) = *(const float4*)(W + i);
  __syncthreads();

  int wave  = tid >> 5;
  int lane  = tid & 31;
  int col   = lane & 15;       // node within 16-tile / A row within tile
  int khalf = lane >> 4;       // 0: K0,K1   1: K2,K3
  int node  = blockIdx.x * 128 + wave * 16 + col;
  int nodeC = node < N ? node : (N - 1);
  bool inb  = node < N;

  // Preload B fragments for all 32 K-steps (K=128, 4 per step)
  v2f breg[32];
  const float* xrow = x + (size_t)nodeC * FDIM + khalf * 2;
#pragma unroll
  for (int s = 0; s < 32; ++s) breg[s] = *(const v2f*)(xrow + s * 4);

  for (int ft = 0; ft < 8; ++ft) {
    v8f acc = {};
    const float* wrow = Wlds + (size_t)(ft * 16 + col) * FDIM + khalf * 2;
#pragma unroll
    for (int s = 0; s < 32; ++s) {
      v2f afrag = *(const v2f*)(wrow + s * 4);
      acc = __builtin_amdgcn_wmma_f32_16x16x4_f32(
          false, afrag, false, breg[s], (short)0, acc, false, false);
    }
    // D layout: VGPR j -> M=j (lanes 0-15), M=j+8 (lanes 16-31); N = lane&15
#pragma unroll
    for (int j = 0; j < 8; ++j) {
      int f = ft * 16 + j + 8 * khalf;
      float vv = acc[j];
      if (inb) hT[(size_t)node * FDIM + f] = vv;
      float sq = inb ? vv * vv : 0.0f;
      sq += __shfl_xor(sq, 1);
      sq += __shfl_xor(sq, 2);
      sq += __shfl_xor(sq, 4);
      sq += __shfl_xor(sq, 8);
      if (col == 0) atomicAdd(&rowsumsq[f], sq);
    }
  }
}

// ---------------------------------------------------------------- row scalar S_f
// S_f folds mobius_matvec scale + proj + logmap0 (all row-wise over N axis).
__global__ void kScal1(const float* __restrict__ W,
                       const float* __restrict__ rowsumsq,
                       float* __restrict__ S) {
  int f = threadIdx.x;  // 128 threads
  float ws = 0.0f;
  for (int k = 0; k < FDIM; ++k) { float w = W[f * FDIM + k]; ws += w * w; }
  float xn  = fmaxf(sqrtf(ws), MIN_NORM);
  float ssq = rowsumsq[f];
  float mxn = fmaxf(sqrtf(ssq), MIN_NORM);
  float t1  = tanhf(mxn / xn * atanhf(fminf(xn, CLIP1)));
  float m1  = t1 / mxn;                        // mobius result scale
  float mobn = fmaxf(t1, MIN_NORM);            // mobius row norm (analytic)
  float sc2 = mobn > MAXNORM ? MAXNORM / mobn : 1.0f;   // proj
  float pn  = fminf(mobn, MAXNORM);
  float slog = atanhf(fminf(pn, CLIP1)) / pn;  // logmap0
  float Sf = m1 * sc2 * slog;
  if (ssq == 0.0f) Sf = 0.0f;                  // all-zero-row case
  S[f] = Sf;
}

// ---------------------------------------------------------------- src/dst histograms
__global__ void kCnt(const int* __restrict__ src, const int* __restrict__ dst,
                     unsigned* __restrict__ cs, unsigned* __restrict__ cd, int E) {
  int i = blockIdx.x * blockDim.x + threadIdx.x;
  if (i < E) {
    atomicAdd(&cs[src[i]], 1u);
    atomicAdd(&cd[dst[i]], 1u);
  }
}

// ---------------------------------------------------------------- hT *= S_f (in place)
__global__ void kScaleH(float* __restrict__ hT, const float* __restrict__ S, size_t total) {
  size_t i = (size_t)blockIdx.x * blockDim.x + threadIdx.x;
  size_t stride = (size_t)gridDim.x * blockDim.x;
  for (; i < total; i += stride) hT[i] *= S[i & (FDIM - 1)];
}

// ---------------------------------------------------------------- count-weighted row sumsq
// Usrc[f] = sum_n cnt_src[n]*h[f,n]^2  (== row sumsq of gathered edge matrix)
#define KU_CHUNK 512
__global__ __launch_bounds__(256) void kU(const float* __restrict__ hT,
                                          const unsigned* __restrict__ cs,
                                          const unsigned* __restrict__ cd,
                                          float* __restrict__ Usrc,
                                          float* __restrict__ Udst, int N) {
  __shared__ float red[512];
  int t = threadIdx.x;
  int f = t & (FDIM - 1), sub = t >> 7;  // 2 node-substreams per f
  int n0 = blockIdx.x * KU_CHUNK;
  int nEnd = min(n0 + KU_CHUNK, N);
  float as_ = 0.0f, ad_ = 0.0f;
  for (int n = n0 + sub; n < nEnd; n += 2) {
    float hv = hT[(size_t)n * FDIM + f];
    float h2 = hv * hv;
    as_ += h2 * (float)cs[n];
    ad_ += h2 * (float)cd[n];
  }
  red[t] = as_; red[256 + t] = ad_;
  __syncthreads();
  if (sub == 0) {
    atomicAdd(&Usrc[f], red[f] + red[128 + f]);
    atomicAdd(&Udst[f], red[256 + f] + red[256 + 128 + f]);
  }
}

// ---------------------------------------------------------------- edge-row scales -> asv/adv
__global__ void kScal2(const float* __restrict__ a, const float* __restrict__ Usrc,
                       const float* __restrict__ Udst, float* __restrict__ asv,
                       float* __restrict__ adv, float* __restrict__ scal) {
  __shared__ float red[128];
  int r = threadIdx.x;  // 128 threads
  float av = a[r], av2 = a[r + FDIM];
  red[r] = av * av + av2 * av2;
  __syncthreads();
  for (int s = 64; s > 0; s >>= 1) { if (r < s) red[r] += red[r + s]; __syncthreads(); }
  if (r == 0) scal[0] = red[0];  // ||a||^2

  // expmap0 + proj, per edge-matrix row (norm over E axis)
  float un  = fmaxf(sqrtf(Usrc[r]), MIN_NORM);
  float e1  = tanhf(un) / un;
  float pn  = fmaxf(tanhf(un), MIN_NORM);
  float sc2 = pn > MAXNORM ? MAXNORM / pn : 1.0f;
  asv[r] = av * e1 * sc2;

  float und  = fmaxf(sqrtf(Udst[r]), MIN_NORM);
  float e1d  = tanhf(und) / und;
  float pnd  = fmaxf(tanhf(und), MIN_NORM);
  float sc2d = pnd > MAXNORM ? MAXNORM / pnd : 1.0f;
  adv[r] = av2 * e1d * sc2d;
}

// ---------------------------------------------------------------- u[n], v[n] (wave per node)
__global__ __launch_bounds__(256) void kUV(const float* __restrict__ hT,
                                           const float* __restrict__ asv,
                                           const float* __restrict__ adv,
                                           float* __restrict__ u, float* __restrict__ v, int N) {
  int wave = threadIdx.x >> 5, lane = threadIdx.x & 31;
  int n = blockIdx.x * 8 + wave;
  if (n >= N) return;
  float us = 0.0f, vs = 0.0f;
  const float* hrow = hT + (size_t)n * FDIM;
#pragma unroll
  for (int t = 0; t < 4; ++t) {
    int f = lane + 32 * t;
    float hv = hrow[f];
    us += asv[f] * hv;
    vs += adv[f] * hv;
  }
  for (int m = 16; m > 0; m >>= 1) { us += __shfl_xor(us, m); vs += __shfl_xor(vs, m); }
  if (lane == 0) { u[n] = us; v[n] = vs; }
}

// ---------------------------------------------------------------- per-edge dot + sumsq
__global__ __launch_bounds__(256) void kEdge1(const int* __restrict__ src,
                                              const int* __restrict__ dst,
                                              const float* __restrict__ u,
                                              const float* __restrict__ v,
                                              float* __restrict__ mxe,
                                              float* __restrict__ scal, int E) {
  __shared__ float red[256];
  int i = blockIdx.x * 256 + threadIdx.x;
  float me = 0.0f; int nz = 0;
  if (i < E) {
    me = u[src[i]] + v[dst[i]];
    mxe[i] = me;
    nz = (me != 0.0f);
  }
  red[threadIdx.x] = me * me;
  __syncthreads();
  for (int s = 128; s > 0; s >>= 1) { if (threadIdx.x < s) red[threadIdx.x] += red[threadIdx.x + s]; __syncthreads(); }
  if (threadIdx.x == 0) atomicAdd(&scal[1], red[0]);
  if ((threadIdx.x & 31) == 0 && __ballot(nz)) atomicOr((int*)scal + 2, 1);
}

// ---------------------------------------------------------------- edge scalar chain
__global__ void kScal3(float* __restrict__ scal) {
  float an  = fmaxf(sqrtf(scal[0]), MIN_NORM);
  float mxn = fmaxf(sqrtf(scal[1]), MIN_NORM);
  float t1  = tanhf(mxn / an * atanhf(fminf(an, CLIP1)));
  float m1  = t1 / mxn;
  float mobn = fmaxf(t1, MIN_NORM);
  float sc2 = mobn > MAXNORM ? MAXNORM / mobn : 1.0f;
  float pn  = fminf(mobn, MAXNORM);
  float slog = atanhf(fminf(pn, CLIP1)) / pn;
  float SE = m1 * sc2 * slog;
  if (((int*)scal)[2] == 0) SE = 0.0f;  // all mx_e == 0
  scal[3] = SE;
}

// ---------------------------------------------------------------- ee = exp(-gelu(SE*mxe))
__global__ void kEdge2(const float* __restrict__ mxe, const float* __restrict__ scal,
                       float* __restrict__ ee, int E) {
  int i = blockIdx.x * blockDim.x + threadIdx.x;
  if (i >= E) return;
  float xe = scal[3] * mxe[i];
  float g = 0.5f * xe * (1.0f + erff(xe * 0.70710678118654752f));
  ee[i] = expf(-g);
}

// ---------------------------------------------------------------- scatter-aggregate
// Wave per DEG-edge group; fast path when all src in the group are equal.
__global__ __launch_bounds__(256) void kAgg(const int* __restrict__ src,
                                            const int* __restrict__ dst,
                                            const float* __restrict__ ee,
                                            const float* __restrict__ hT,
                                            float* __restrict__ hp,
                                            float* __restrict__ rowsum,
                                            int E, int DEG) {
  int wave = threadIdx.x >> 5, lane = threadIdx.x & 31;
  long g = (long)blockIdx.x * 8 + wave;
  long base = g * DEG;
  if (base >= E) return;
  int cnt = (int)min((long)DEG, (long)E - base);
  int s0 = src[base];
  bool uni = true;
  for (int d = 1; d < cnt; ++d) uni = uni && (src[base + d] == s0);
  if (uni) {
    float r0 = 0, r1 = 0, r2 = 0, r3 = 0, esum = 0;
    for (int d = 0; d < cnt; ++d) {
      long e = base + d;
      float w = ee[e];
      const float* hcol = hT + (size_t)dst[e] * FDIM;
      r0 += w * hcol[lane];       r1 += w * hcol[lane + 32];
      r2 += w * hcol[lane + 64];  r3 += w * hcol[lane + 96];
      esum += w;
    }
    float* hrow = hp + (size_t)s0 * FDIM;
    atomicAdd(&hrow[lane], r0);      atomicAdd(&hrow[lane + 32], r1);
    atomicAdd(&hrow[lane + 64], r2); atomicAdd(&hrow[lane + 96], r3);
    if (lane == 0) atomicAdd(&rowsum[s0], esum);
  } else {
    for (int d = 0; d < cnt; ++d) {
      long e = base + d;
      float w = ee[e];
      const float* hcol = hT + (size_t)dst[e] * FDIM;
      float* hrow = hp + (size_t)src[e] * FDIM;
      atomicAdd(&hrow[lane],      w * hcol[lane]);
      atomicAdd(&hrow[lane + 32], w * hcol[lane + 32]);
      atomicAdd(&hrow[lane + 64], w * hcol[lane + 64]);
      atomicAdd(&hrow[lane + 96], w * hcol[lane + 96]);
      if (lane == 0) atomicAdd(&rowsum[src[e]], w);
    }
  }
}

// ---------------------------------------------------------------- elu + expmap0 + proj
__global__ __launch_bounds__(256) void kFinal(const float* __restrict__ hp,
                                              const float* __restrict__ rowsum,
                                              float* __restrict__ out, int N) {
  int wave = threadIdx.x >> 5, lane = threadIdx.x & 31;
  int n = blockIdx.x * 8 + wave;
  if (n >= N) return;
  float inv = 1.0f / rowsum[n];
  float y[4]; float ssq = 0.0f;
#pragma unroll
  for (int t = 0; t < 4; ++t) {
    float hv = hp[(size_t)n * FDIM + lane + 32 * t] * inv;
    float e = hv > 0.0f ? hv : expm1f(hv);   // elu
    y[t] = e; ssq += e * e;
  }
  for (int m = 16; m > 0; m >>= 1) ssq += __shfl_xor(ssq, m);
  float un = fmaxf(sqrtf(ssq), MIN_NORM);
  float e1 = tanhf(un) / un;                   // expmap0
  float pn = fmaxf(tanhf(un), MIN_NORM);
  float sc2 = pn > MAXNORM ? MAXNORM / pn : 1.0f;  // proj
  float sc = e1 * sc2;
#pragma unroll
  for (int t = 0; t < 4; ++t) out[(size_t)n * FDIM + lane + 32 * t] = y[t] * sc;
}

// ================================================================ launch
extern "C" void kernel_launch(void* const* d_in, const int* in_sizes, int n_in,
                              void* d_out, int out_size, void* d_ws, size_t ws_size,
                              hipStream_t stream) {
  const float* x    = (const float*)d_in[0];
  const int*   edge = (const int*)d_in[1];
  const float* W    = (const float*)d_in[2];
  const float* a    = (const float*)d_in[3];
  float* out = (float*)d_out;

  const size_t N = (size_t)in_sizes[0] / FDIM;
  const size_t E = (size_t)in_sizes[1] / 2;
  const int DEG = (int)((E + N - 1) / N);
  const int* src = edge;
  const int* dst = edge + E;

  // workspace layout (float units)
  float* ws = (float*)d_ws;
  size_t o = 0;
  float* hT   = ws + o; o += N * FDIM;   // xW then h (node-major)
  float* u    = ws + o; o += N;
  float* v    = ws + o; o += N;
  float* mxe  = ws + o; o += E;
  float* ee   = ws + o; o += E;
  size_t zeroStart = o;
  float* hp   = ws + o; o += N * FDIM;   // aggregated numerator
  float* rs   = ws + o; o += N;          // rowsum
  unsigned* cs = (unsigned*)(ws + o); o += N;
  unsigned* cd = (unsigned*)(ws + o); o += N;
  float* rss  = ws + o; o += FDIM;       // rowsumsq of mx
  float* S    = ws + o; o += FDIM;
  float* Usrc = ws + o; o += FDIM;
  float* Udst = ws + o; o += FDIM;
  float* asv  = ws + o; o += FDIM;
  float* adv  = ws + o; o += FDIM;
  float* scal = ws + o; o += 16;
  size_t zeroLen = o - zeroStart;

  // 1) zero accumulators
  kZero<<<2048, 256, 0, stream>>>(ws + zeroStart, zeroLen);
  // 2) WMMA GEMM + row sumsq
  kGemm<<<(int)((N + 127) / 128), 256, 0, stream>>>(x, W, hT, rss, (int)N);
  // 3) per-row scalars
  kScal1<<<1, FDIM, 0, stream>>>(W, rss, S);
  // 4) histograms
  kCnt<<<(int)((E + 255) / 256), 256, 0, stream>>>(src, dst, cs, cd, (int)E);
  // 5) h = mx * S (in place)
  kScaleH<<<4096, 256, 0, stream>>>(hT, S, N * FDIM);
  // 6) count-weighted row sumsq of h
  kU<<<(int)((N + KU_CHUNK - 1) / KU_CHUNK), 256, 0, stream>>>(hT, cs, cd, Usrc, Udst, (int)N);
  // 7) edge-row scale vectors
  kScal2<<<1, FDIM, 0, stream>>>(a, Usrc, Udst, asv, adv, scal);
  // 8) u/v per-node dot products
  kUV<<<(int)((N + 7) / 8), 256, 0, stream>>>(hT, asv, adv, u, v, (int)N);
  // 9) per-edge attention logits + global sumsq
  kEdge1<<<(int)((E + 255) / 256), 256, 0, stream>>>(src, dst, u, v, mxe, scal, (int)E);
  // 10) scalar chain for edge vector
  kScal3<<<1, 1, 0, stream>>>(scal);
  // 11) ee = exp(-gelu(.))
  kEdge2<<<(int)((E + 255) / 256), 256, 0, stream>>>(mxe, scal, ee, (int)E);
  // 12) scatter aggregation
  {
    long groups = (long)((E + DEG - 1) / DEG);
    kAgg<<<(int)((groups + 7) / 8), 256, 0, stream>>>(src, dst, ee, hT, hp, rs, (int)E, DEG);
  }
  // 13) normalize + elu + expmap0 + proj
  kFinal<<<(int)((N + 7) / 8), 256, 0, stream>>>(hp, rs, out, (int)N);
}